// FinePreprocess_28278064677442
// MI455X (gfx1250) — compile-verified
//
#include <hip/hip_runtime.h>

typedef __attribute__((ext_vector_type(2))) float v2f;
typedef __attribute__((ext_vector_type(8))) float v8f;

#define M_MATCH 5000
#define L_COARSE 4800
#define CF 128
#define CC 256
#define HF_DIM 240
#define WF_DIM 320
#define WW 25
#define SFS 132  // padded LDS row stride (floats): banks (4*ln16+k)%64, conflict-free

// Repack weights once:
//  BW : W1 (fine half of merge_w) in WMMA B-fragment order:
//       BW[((wave*32 + ks)*32 + lane)*2 + p] = merge_w[n][k],
//       n = wave*16 + (lane&15), k = ks*4 + (lane>>4)*2 + p
//       -> hot loop does one coalesced global_load_b64 per lane per k-step.
//  W2T[k*128+n] = merge_w[n][128+k]
//  DPT[k*128+n] = down_proj_w[n][k]
__global__ void prep_weights_kernel(const float* __restrict__ merge_w,
                                    const float* __restrict__ down_proj_w,
                                    float* __restrict__ BW,
                                    float* __restrict__ W2T,
                                    float* __restrict__ DPT) {
  int idx = blockIdx.x * blockDim.x + threadIdx.x;
  if (idx < 128 * 128) {
    int p    = idx & 1;
    int lane = (idx >> 1) & 31;
    int ks   = (idx >> 6) & 31;
    int wave = idx >> 11;
    int n = wave * 16 + (lane & 15);
    int k = ks * 4 + ((lane >> 4) << 1) + p;
    BW[idx] = merge_w[n * 256 + k];

    int kk = idx >> 7, nn = idx & 127;
    W2T[idx] = merge_w[nn * 256 + 128 + kk];
  }
  if (idx < 256 * 128) {
    int kk = idx >> 7, nn = idx & 127;
    DPT[idx] = down_proj_w[nn * 256 + kk];
  }
}

// One block per match-side s in [0, 2M). 256 threads = 8 wave32.
__global__ __launch_bounds__(256)
void fine_preprocess_kernel(const float* __restrict__ feat_f0,
                            const float* __restrict__ feat_f1,
                            const float* __restrict__ feat_c0,
                            const float* __restrict__ feat_c1,
                            const int* __restrict__ b_ids,
                            const int* __restrict__ i_ids,
                            const int* __restrict__ j_ids,
                            const float* __restrict__ down_proj_b,
                            const float* __restrict__ merge_b,
                            const float* __restrict__ BW,
                            const float* __restrict__ W2T,
                            const float* __restrict__ DPT,
                            float* __restrict__ out) {
  __shared__ float sF[32 * SFS];  // gathered fine window (padded rows), 25..31 zero
  __shared__ float sC[256];       // coarse feature row
  __shared__ float sDP[128];      // down-projected coarse
  __shared__ float sT[128];       // per-match constant: dp @ W2^T + merge_b

  const int tid = threadIdx.x;
  const int s = blockIdx.x;
  const int m = (s < M_MATCH) ? s : s - M_MATCH;
  const int b = b_ids[m];
  const int l = (s < M_MATCH) ? i_ids[m] : j_ids[m];
  const float* __restrict__ ff = (s < M_MATCH) ? feat_f0 : feat_f1;
  const float* __restrict__ fc = (s < M_MATCH) ? feat_c0 : feat_c1;

  sC[tid] = fc[((size_t)b * L_COARSE + l) * CC + tid];

  // Gather: invert unfold(K=5,stride=4,pad=2) composed with the flat .view.
  // flat = l*3200 + w*128 + ch  ==  c*120000 + kk*4800 + pos  (per batch)
  for (int i = tid; i < 32 * 128; i += 256) {
    int row = i >> 7;        // window position (0..24), 25..31 padded
    int ch  = i & 127;
    float v = 0.0f;
    if (row < WW) {
      int flat = l * (WW * CF) + row * CF + ch;
      int c    = flat / 120000;
      int rem  = flat - c * 120000;
      int kk   = rem / 4800;
      int pos  = rem - kk * 4800;
      int kh   = kk / 5;
      int kw   = kk - kh * 5;
      int ho   = pos / 80;
      int wo   = pos - ho * 80;
      int h  = ho * 4 + kh - 2;
      int w  = wo * 4 + kw - 2;
      if ((unsigned)h < (unsigned)HF_DIM && (unsigned)w < (unsigned)WF_DIM)
        v = ff[(((size_t)b * CF + c) * HF_DIM + h) * WF_DIM + w];
    }
    sF[row * SFS + ch] = v;
  }
  __syncthreads();

  // dp = coarse_row @ down_proj_w^T + down_proj_b   (VALU; negligible work)
  if (tid < 128) {
    float acc = down_proj_b[tid];
#pragma unroll 8
    for (int k = 0; k < 256; ++k) acc += sC[k] * DPT[k * 128 + tid];
    sDP[tid] = acc;
  }
  __syncthreads();
  // t = dp @ W2^T + merge_b  (constant over the 25 window rows)
  if (tid < 128) {
    float acc = merge_b[tid];
#pragma unroll 8
    for (int k = 0; k < 128; ++k) acc += sDP[k] * W2T[k * 128 + tid];
    sT[tid] = acc;
  }
  __syncthreads();

  // WMMA GEMM: D[32x128] = sF[32x128] @ W1^T, wave w owns N-tile w.
  const int wave = tid >> 5;
  const int lane = tid & 31;
  const int half = lane >> 4;   // lanes 16..31 carry K+2 / M+8 halves
  const int ln16 = lane & 15;
  const int n0   = wave * 16;
  const int col  = n0 + ln16;

  // B fragments pre-permuted: one b64 per lane per k-step, lane-coalesced.
  const v2f* __restrict__ bw = (const v2f*)BW + ((size_t)wave * 32) * 32 + lane;

  v8f acc0 = {};  // rows 0..15
  v8f acc1 = {};  // rows 16..31 (only 16..24 stored)
#pragma unroll
  for (int ks = 0; ks < 32; ++ks) {
    const int kA = ks * 4 + half * 2;
    v2f a0, a1;
    a0.x = sF[ln16 * SFS + kA];
    a0.y = sF[ln16 * SFS + kA + 1];
    a1.x = sF[(16 + ln16) * SFS + kA];
    a1.y = sF[(16 + ln16) * SFS + kA + 1];
    v2f bf = bw[ks * 32];
    acc0 = __builtin_amdgcn_wmma_f32_16x16x4_f32(false, a0, false, bf,
                                                 (short)0, acc0, false, false);
    acc1 = __builtin_amdgcn_wmma_f32_16x16x4_f32(false, a1, false, bf,
                                                 (short)0, acc1, false, false);
  }

  // D layout (16x16 f32): VGPR v -> M = v + half*8, N = col.
  const float tcol = sT[col];
  const size_t base = (size_t)s * (WW * CF);
#pragma unroll
  for (int v = 0; v < 8; ++v) {
    int row = v + half * 8;
    out[base + (size_t)row * CF + col] = acc0[v] + tcol;
    int r1 = row + 16;
    if (r1 < WW) out[base + (size_t)r1 * CF + col] = acc1[v] + tcol;
  }
}

extern "C" void kernel_launch(void* const* d_in, const int* in_sizes, int n_in,
                              void* d_out, int out_size, void* d_ws, size_t ws_size,
                              hipStream_t stream) {
  const float* feat_f0     = (const float*)d_in[0];
  const float* feat_f1     = (const float*)d_in[1];
  const float* feat_c0     = (const float*)d_in[2];
  const float* feat_c1     = (const float*)d_in[3];
  const int*   b_ids       = (const int*)d_in[4];
  const int*   i_ids       = (const int*)d_in[5];
  const int*   j_ids       = (const int*)d_in[6];
  const float* down_proj_w = (const float*)d_in[7];
  const float* down_proj_b = (const float*)d_in[8];
  const float* merge_w     = (const float*)d_in[9];
  const float* merge_b     = (const float*)d_in[10];

  float* BW  = (float*)d_ws;              // 128*128 (W1 in fragment order)
  float* W2T = BW + 128 * 128;            // 128*128
  float* DPT = W2T + 128 * 128;           // 256*128  (total 256 KB of ws)

  prep_weights_kernel<<<128, 256, 0, stream>>>(merge_w, down_proj_w, BW, W2T, DPT);
  fine_preprocess_kernel<<<2 * M_MATCH, 256, 0, stream>>>(
      feat_f0, feat_f1, feat_c0, feat_c1, b_ids, i_ids, j_ids,
      down_proj_b, merge_b, BW, W2T, DPT, (float*)d_out);
}